// QuantizationLayer_35708358099464
// MI455X (gfx1250) — compile-verified
//
#include <hip/hip_runtime.h>
#include <math.h>

// CDNA5 / gfx1250: wave32, streaming-bound elementwise SAR-ADC quantizer.
// Strategy: b128 NT loads/stores, global_prefetch_b8 ahead of the stride,
// hardware v_tanh_f32 transcendental (gfx1250), grid-stride loop,
// 4 elements (=1 float4 of x, 1 float4 of q, 3 float4 of Q) per iteration.

typedef float __attribute__((ext_vector_type(4))) v4f;

#define Q_VREF  0.225f     // VDD / 2^3
#define Q_VR    0.225f
#define Q_AMP   10000.0f
#define Q_DELTA 1e-30f

// gfx1250 hardware tanh (TRANS op, quarter-rate, co-executes with VALU).
// Falls back to OCML tanhf if the builtin is not available in this toolchain.
#if defined(__has_builtin)
#if __has_builtin(__builtin_amdgcn_tanhf)
#define Q_TANH(v) __builtin_amdgcn_tanhf(v)
#endif
#endif
#ifndef Q_TANH
#define Q_TANH(v) tanhf(v)
#endif

// One element through the 3-bit successive-approximation chain.
// Association mirrors the JAX reference exactly (fp contract off so the
// compiler doesn't re-round via FMA): bit_sum = base + ((Q+1)*0.5)*Wn*VR ...
__device__ __forceinline__ void adc3(float xv,
                                     float wn0, float wn1, float wn2,
                                     float wn3, float wn4, float wn5,
                                     float& Q0, float& Q1, float& Q2, float& qv)
{
#pragma clang fp contract(off)
    float bs2 = wn5 * Q_VREF;
    Q2 = Q_TANH(Q_AMP * (xv - bs2 + Q_DELTA));

    float bs1 = wn4 * Q_VREF + ((Q2 + 1.0f) * 0.5f) * wn3 * Q_VR;
    Q1 = Q_TANH(Q_AMP * (xv - bs1 + Q_DELTA));

    float bs0 = wn2 * Q_VREF + ((Q1 + 1.0f) * 0.5f) * wn1 * Q_VR
                             + ((Q2 + 1.0f) * 0.5f) * wn0 * Q_VR;
    Q0 = Q_TANH(Q_AMP * (xv - bs0 + Q_DELTA));

    // bit_weights = [2^b * VR * 0.5] = [0.1125, 0.225, 0.45]
    qv = (Q0 + 1.0f) * 0.1125f + (Q1 + 1.0f) * 0.225f + (Q2 + 1.0f) * 0.45f;
}

__global__ void __launch_bounds__(256)
quant_sar_kernel(const float* __restrict__ x,
                 const float* __restrict__ W,
                 const float* __restrict__ noise,
                 float* __restrict__ qout,   // [N]
                 float* __restrict__ Qout,   // [N,3]
                 float* __restrict__ wnout,  // [6]
                 long long n4,               // N/4 vec groups
                 long long ntotal)           // N
{
    // Tail output: noisy weights (6 scalars), written once per launch.
    if (blockIdx.x == 0 && threadIdx.x < 6) {
        wnout[threadIdx.x] = W[threadIdx.x] + noise[threadIdx.x];
    }

    // Uniform scalars: invariant addresses -> scalar loads (s_load_b128).
    const float wn0 = W[0] + noise[0];
    const float wn1 = W[1] + noise[1];
    const float wn2 = W[2] + noise[2];
    const float wn3 = W[3] + noise[3];
    const float wn4 = W[4] + noise[4];
    const float wn5 = W[5] + noise[5];

    const long long stride = (long long)gridDim.x * blockDim.x;
    long long i = (long long)blockIdx.x * blockDim.x + threadIdx.x;

    for (; i < n4; i += stride) {
        // gfx1250 prefetch of the next tile this thread will touch.
        if (i + stride < n4) {
            __builtin_prefetch(x + 4 * (i + stride), 0, 0);
        }

        // Single-touch streaming data: non-temporal b128 load.
        v4f xv = __builtin_nontemporal_load((const v4f*)(x + 4 * i));

        float q0a, q1a, q2a, qa;  adc3(xv.x, wn0, wn1, wn2, wn3, wn4, wn5, q0a, q1a, q2a, qa);
        float q0b, q1b, q2b, qb;  adc3(xv.y, wn0, wn1, wn2, wn3, wn4, wn5, q0b, q1b, q2b, qb);
        float q0c, q1c, q2c, qc;  adc3(xv.z, wn0, wn1, wn2, wn3, wn4, wn5, q0c, q1c, q2c, qc);
        float q0d, q1d, q2d, qd;  adc3(xv.w, wn0, wn1, wn2, wn3, wn4, wn5, q0d, q1d, q2d, qd);

        // q: contiguous float4, NT store (no reuse of output lines).
        v4f qv = { qa, qb, qc, qd };
        __builtin_nontemporal_store(qv, (v4f*)(qout + 4 * i));

        // Q: [elem][bit] layout -> 12 contiguous floats = 3 aligned b128 stores.
        float* Qp = Qout + 12 * i;
        v4f t0 = { q0a, q1a, q2a, q0b };
        v4f t1 = { q1b, q2b, q0c, q1c };
        v4f t2 = { q2c, q0d, q1d, q2d };
        __builtin_nontemporal_store(t0, (v4f*)(Qp + 0));
        __builtin_nontemporal_store(t1, (v4f*)(Qp + 4));
        __builtin_nontemporal_store(t2, (v4f*)(Qp + 8));
    }

    // Scalar tail for N % 4 != 0 (not hit for the reference shape).
    for (long long idx = n4 * 4 + (long long)blockIdx.x * blockDim.x + threadIdx.x;
         idx < ntotal; idx += stride) {
        float Q0, Q1, Q2, qs;
        adc3(x[idx], wn0, wn1, wn2, wn3, wn4, wn5, Q0, Q1, Q2, qs);
        qout[idx]         = qs;
        Qout[3 * idx]     = Q0;
        Qout[3 * idx + 1] = Q1;
        Qout[3 * idx + 2] = Q2;
    }
}

extern "C" void kernel_launch(void* const* d_in, const int* in_sizes, int n_in,
                              void* d_out, int out_size, void* d_ws, size_t ws_size,
                              hipStream_t stream) {
    (void)n_in; (void)out_size; (void)d_ws; (void)ws_size;

    const float* x     = (const float*)d_in[0];  // [L*A] = [1048576*24]
    const float* W     = (const float*)d_in[1];  // [6]
    const float* noise = (const float*)d_in[2];  // [6]

    const long long N  = (long long)in_sizes[0];
    const long long n4 = N / 4;

    float* out   = (float*)d_out;
    float* qout  = out;            // N floats
    float* Qout  = out + N;        // 3N floats
    float* wnout = out + 4 * N;    // 6 floats

    const int threads = 256;                 // 8 wave32 per block
    const long long iters_per_thread = 4;    // prefetch distance = 1 stride
    long long want_threads = (n4 + iters_per_thread - 1) / iters_per_thread;
    if (want_threads < threads) want_threads = threads;
    int blocks = (int)((want_threads + threads - 1) / threads);  // 6144 nominal

    quant_sar_kernel<<<blocks, threads, 0, stream>>>(
        x, W, noise, qout, Qout, wnout, n4, N);
}